// BDRFuse_54305566490942
// MI455X (gfx1250) — compile-verified
//
#include <hip/hip_runtime.h>
#include <hip/hip_bf16.h>
#include <math.h>

typedef float v2f __attribute__((ext_vector_type(2)));
typedef float v8f __attribute__((ext_vector_type(8)));

#define Bx 8
#define Tt 64
#define Nn 196
#define Dd 768
#define Kk 8
#define FEAT 40           // K*(P+3)
#define BOUNDF 2.5f
#define EPSF 1e-6f
#define XSTR 772          // 768 + 4 pad (bank stagger)
#define PI_F 3.14159265358979323846f

__device__ __forceinline__ void async_wait0() {
#if __has_builtin(__builtin_amdgcn_s_wait_asynccnt)
    __builtin_amdgcn_s_wait_asynccnt(0);
#else
    asm volatile("s_wait_asynccnt 0" ::: "memory");
#endif
}

// ---------------- kernel 0: W = softplus(W_raw), write to d_out tail + scratch ----------
__global__ __launch_bounds__(256) void softplus_kernel(const float* __restrict__ W_raw,
                                                       float* __restrict__ Wsp,
                                                       float* __restrict__ Wout, int n) {
    int i = blockIdx.x * 256 + threadIdx.x;
    if (i < n) {
        float w = W_raw[i];
        float sp = (w > 20.0f) ? w : log1pf(expf(w));
        Wsp[i]  = sp;
        Wout[i] = sp;
    }
}

// ---------------- kernel 1: fully fused per-(b,n) pipeline -----------------------------
// One workgroup per (b,n). x[b,:,n,:] (64x768 fp32 = 192KB) staged once in LDS via
// GLOBAL_LOAD_ASYNC_TO_LDS_B128, projection via v_wmma_f32_16x16x4_f32 fed by
// immediate-offset ds_load_b64, temporal stats + DCT + GEMV in-LDS, h written once.
__global__ __launch_bounds__(256) void fused_kernel(const float* __restrict__ x,
                                                    const float* __restrict__ valid_mask,
                                                    const float* __restrict__ proj_W,
                                                    const float* __restrict__ Wsp,
                                                    const float* __restrict__ beta,
                                                    float* __restrict__ h) {
    __shared__ __align__(16) float xs[Tt * XSTR];   // 197,632 B  x tile [t][d]
    __shared__ __align__(16) float Bpk[192 * 64];   //  49,152 B  WMMA B operands, lane-packed
    __shared__ float vbuf[2][Kk][Tt];               //   4,096 B  projection partials per D-half
    __shared__ float vb[Kk][Tt];                    //   2,048 B  bounded traces
    __shared__ float rmsk[Kk];
    __shared__ float feats[FEAT];
    __shared__ float ys[Dd];                        //   3,072 B  sigmoid(beta)*y
    __shared__ float msk[Tt];

    const int tid = threadIdx.x;
    const int blk = blockIdx.x;
    const int b = blk / Nn;
    const int n = blk % Nn;

    // ---- stage x tile: async global->LDS b128, no VGPR round-trip (ASYNCcnt path) ----
    {
        const size_t rowstride = (size_t)Nn * Dd;
        const size_t base = ((size_t)b * Tt) * rowstride + (size_t)n * Dd;
        for (int i4 = tid; i4 < Tt * (Dd / 4); i4 += 256) {
            int t = i4 / (Dd / 4);
            int c = (i4 % (Dd / 4)) * 4;
            const float* gp = x + base + (size_t)t * rowstride + c;
            unsigned int ldsoff = (unsigned int)(uintptr_t)&xs[t * XSTR + c];
            asm volatile("global_load_async_to_lds_b128 %0, %1, off"
                         :: "v"(ldsoff), "v"(gp) : "memory");
        }
    }

    // ---- pack B operands: Bpk[s][lane][j] = U[s*4 + row][col], zero for col>=8 ----
    //      row = (lane<16 ? j : 2+j), col = lane&15;  U[d][k] = proj_W[k*768+d]
    for (int e = tid; e < 192 * 64; e += 256) {
        int s    = e >> 6;
        int r    = e & 63;
        int lane = r >> 1;
        int j    = r & 1;
        int row  = (lane < 16) ? j : (2 + j);
        int col  = lane & 15;
        Bpk[e] = (col < Kk) ? proj_W[col * Dd + s * 4 + row] : 0.0f;
    }
    if (tid < Tt) msk[tid] = valid_mask[((size_t)b * Tt + tid) * Nn + n];

    async_wait0();
    __syncthreads();

    // ---- projection: v[t][k] = sum_d xs[t][d] * U[d][k]  via WMMA f32 16x16x4 ----
    // wave w: t-tile = w&3 (16 rows), D-half = w>>2 (384 d's -> 96 wmma steps)
    {
        const int wave = tid >> 5;
        const int lane = tid & 31;
        const int tile = wave & 3;
        const int half = wave >> 2;
        const int mrow = lane & 15;
        const bool lo  = lane < 16;
        const int abase = (tile * 16 + mrow) * XSTR + half * 384 + (lo ? 0 : 2);
        const float* bbase = &Bpk[(half * 96) * 64 + lane * 2];

        v8f acc = {};
        #pragma unroll 8
        for (int s = 0; s < 96; ++s) {
            v2f a  = *(const v2f*)&xs[abase + s * 4];     // A 16x4: lanes0-15 K=0/1, 16-31 K=2/3
            v2f bb = *(const v2f*)&bbase[s * 64];          // pre-packed, ds_load_b64, imm offset
            acc = __builtin_amdgcn_wmma_f32_16x16x4_f32(
                false, a, false, bb, (short)0, acc, false, false);
        }
        // D 16x16 f32: VGPR i -> row M = i + (lane<16?0:8), col N = lane%16
        if (mrow < Kk) {
            #pragma unroll
            for (int i = 0; i < 8; ++i) {
                int trow = tile * 16 + i + (lo ? 0 : 8);
                vbuf[half][mrow][trow] = acc[i];
            }
        }
    }
    __syncthreads();

    // ---- combine D-halves + mask ----
    for (int i = tid; i < Kk * Tt; i += 256) {
        int k = i / Tt, t = i % Tt;
        vbuf[0][k][t] = (vbuf[0][k][t] + vbuf[1][k][t]) * msk[t];
    }
    __syncthreads();

    // ---- per-trace RMS ----
    if (tid < Kk) {
        float ss = 0.0f;
        for (int t = 0; t < Tt; ++t) { float v = vbuf[0][tid][t]; ss += v * v; }
        rmsk[tid] = sqrtf(ss * (1.0f / Tt) + EPSF);
    }
    __syncthreads();

    // ---- bounded traces ----
    for (int i = tid; i < Kk * Tt; i += 256) {
        int k = i / Tt, t = i % Tt;
        vb[k][t] = BOUNDF * tanhf(vbuf[0][k][t] / (rmsk[k] + EPSF));
    }
    __syncthreads();

    // ---- 40 features: 3 DCT coeffs + mean + rms per trace ----
    if (tid < FEAT) {
        int k = tid / 5, s = tid % 5;
        float r = 0.0f;
        if (s < 3) {
            float norm;
            if (s == 0) norm = sqrtf((float)Tt) + EPSF;
            else {
                float ss = 0.0f;
                for (int t = 0; t < Tt; ++t) {
                    float c = cosf(PI_F * (float)s * ((float)t + 0.5f) / (float)Tt);
                    ss += c * c;
                }
                norm = sqrtf(ss) + EPSF;
            }
            for (int t = 0; t < Tt; ++t) {
                float c = (s == 0) ? 1.0f
                                   : cosf(PI_F * (float)s * ((float)t + 0.5f) / (float)Tt);
                r += vb[k][t] * (c / norm);
            }
        } else if (s == 3) {
            for (int t = 0; t < Tt; ++t) r += vb[k][t];
            r *= (1.0f / Tt);
        } else {
            for (int t = 0; t < Tt; ++t) r += vb[k][t] * vb[k][t];
            r = sqrtf(r * (1.0f / Tt) + EPSF);
        }
        feats[tid] = r;
    }
    __syncthreads();

    // ---- y[d] = feats . Wsp[:,d], gate by sigmoid(beta) (GEMV; Wsp stays in L2) ----
    for (int d = tid; d < Dd; d += 256) {
        float acc = 0.0f;
        #pragma unroll
        for (int f = 0; f < FEAT; ++f) acc += feats[f] * Wsp[f * Dd + d];
        float bs = 1.0f / (1.0f + expf(-beta[d]));
        ys[d] = bs * acc;
    }
    __syncthreads();

    // ---- h = x + mask*sigmoid(beta)*y  (m*(x+bs*y)+(1-m)*x == x+m*bs*y) ----
    {
        const size_t rowstride = (size_t)Nn * Dd;
        const size_t base = ((size_t)b * Tt) * rowstride + (size_t)n * Dd;
        for (int i4 = tid; i4 < Tt * (Dd / 4); i4 += 256) {
            int t = i4 / (Dd / 4);
            int c = (i4 % (Dd / 4)) * 4;
            float m = msk[t];
            float4 xv = *(const float4*)&xs[t * XSTR + c];
            float4 o;
            o.x = xv.x + m * ys[c + 0];
            o.y = xv.y + m * ys[c + 1];
            o.z = xv.z + m * ys[c + 2];
            o.w = xv.w + m * ys[c + 3];
            *(float4*)(h + base + (size_t)t * rowstride + c) = o;
        }
    }
}

extern "C" void kernel_launch(void* const* d_in, const int* in_sizes, int n_in,
                              void* d_out, int out_size, void* d_ws, size_t ws_size,
                              hipStream_t stream) {
    const float* x          = (const float*)d_in[0];
    const float* valid_mask = (const float*)d_in[1];
    const float* proj_W     = (const float*)d_in[2];
    const float* W_raw      = (const float*)d_in[3];
    const float* beta       = (const float*)d_in[4];

    float* out  = (float*)d_out;
    float* hptr = out;                                         // [B,T,N,D]
    float* Wout = out + (size_t)Bx * Tt * Nn * Dd;             // [40, 768]
    float* Wsp  = (float*)d_ws;                                // scratch softplus(W)

    const int wN = FEAT * Dd;                                  // 30720
    softplus_kernel<<<(wN + 255) / 256, 256, 0, stream>>>(W_raw, Wsp, Wout, wN);
    fused_kernel<<<Bx * Nn, 256, 0, stream>>>(x, valid_mask, proj_W, Wsp, beta, hptr);
}